// RedLayer_22651657519762
// MI455X (gfx1250) — compile-verified
//
#include <hip/hip_runtime.h>
#include <stdint.h>

typedef float v2f __attribute__((ext_vector_type(2)));
typedef float v8f __attribute__((ext_vector_type(8)));

#define B_ROWS 4096
#define F_DIM  2048
#define R_DIM  8
#define O_DIM  2048
#define K_DIM  (F_DIM * R_DIM)   // 16384

// ---- workspace layout ----
#define HIST_BINS  65536
#define HIST_BYTES (HIST_BINS * 4)
#define SEL_OFF    HIST_BYTES            // 8 uints: sel1=[0..3], sel2=[4..7]
#define W1_OFF     (HIST_BYTES + 256)    // float[16384], transposed [F][R]

// ---- GEMM tiling ----
#define BM 128
#define BN 128
#define BK 32
#define KF 32                    // f-columns of x per chunk (=> 256 k, 8 stages)
#define NSTAGE (K_DIM / BK)      // 512
#define NCHUNK (F_DIM / KF)      // 64
#define AS_P 144                 // float2 col stride: 2*144 % 64 == 32 -> kpair halves on disjoint banks
#define BS_P 144
#define XS_STRIDE 129            // 129 % 64 == 1 -> conflict-free transpose fill
#define AS_HALF ((BK / 2) * AS_P)
#define BS_HALF ((BK / 2) * BS_P)
#define XS_HALF (KF * XS_STRIDE)

static __device__ __forceinline__ unsigned abs_key(float v) {
    return __float_as_uint(v) & 0x7fffffffu;
}

// ---------------- radix-select pass 1: histogram of top 16 key bits --------
__global__ void k_hist_hi(const float* __restrict__ s, long n,
                          unsigned* __restrict__ hist) {
    long i = (long)blockIdx.x * blockDim.x + threadIdx.x;
    long stride = (long)gridDim.x * blockDim.x;
    for (; i < n; i += stride) {
        atomicAdd(&hist[abs_key(s[i]) >> 16], 1u);
    }
}

// ---------------- radix-select pass 2: histogram of low 16 bits, filtered --
__global__ void k_hist_lo(const float* __restrict__ s, long n,
                          unsigned* __restrict__ hist,
                          const unsigned* __restrict__ sel) {
    const unsigned hi = sel[0];
    long i = (long)blockIdx.x * blockDim.x + threadIdx.x;
    long stride = (long)gridDim.x * blockDim.x;
    for (; i < n; i += stride) {
        unsigned key = abs_key(s[i]);
        if ((key >> 16) == hi) atomicAdd(&hist[key & 0xffffu], 1u);
    }
}

// ---- find bin containing the k-th smallest (1-based). mode 0: k from imm,
// ---- writes sel[0]=hi bin, sel[1]=remaining rank. mode 1: k from sel[1],
// ---- writes sel[2]=full 32-bit threshold key.
__global__ void k_scan(const unsigned* __restrict__ hist,
                       unsigned* __restrict__ sel, int mode, unsigned k_imm) {
    __shared__ unsigned part[256];
    const int t = threadIdx.x;
    unsigned ssum = 0;
    for (int i = 0; i < 256; ++i) ssum += hist[t * 256 + i];
    part[t] = ssum;
    __syncthreads();
    if (t == 0) {
        unsigned k = (mode == 0) ? k_imm : sel[1];
        unsigned acc = 0;
        int seg = 255;
        for (int i = 0; i < 256; ++i) {
            if (acc + part[i] >= k) { seg = i; break; }
            acc += part[i];
        }
        unsigned bin = (unsigned)(seg * 256 + 255);
        for (int i = 0; i < 256; ++i) {
            unsigned h = hist[seg * 256 + i];
            if (acc + h >= k) { bin = (unsigned)(seg * 256 + i); break; }
            acc += h;
        }
        if (mode == 0) { sel[0] = bin; sel[1] = k - acc; }
        else           { sel[2] = (sel[0] << 16) | bin; }
    }
}

// ---- build masked w1, transposed to [F][R] so the GEMM can float4-load it --
__global__ void k_mask_w1(const float* __restrict__ w1,
                          const float* __restrict__ s1,
                          const unsigned* __restrict__ sel,
                          float* __restrict__ w1mt) {
    int i = blockIdx.x * 256 + threadIdx.x;   // i = f*8 + r
    if (i >= R_DIM * F_DIM) return;
    int f = i >> 3, r = i & 7;
    float w = w1[r * F_DIM + f];
    w1mt[i] = (abs_key(s1[r * F_DIM + f]) >= sel[2]) ? w : 0.0f;
}

// ---------------- fused GEMM: out = relu(x ⊗ w1m) @ (w2*mask2)^T -----------
// Double-buffered pipeline (CDNA5: 320KB LDS/WGP, ~104KB static used here):
// iteration s fills stage s+1 (A from Xs, B from HBM) while WMMAs run stage s.
__global__ __launch_bounds__(256) void k_gemm(
    const float* __restrict__ x, const float* __restrict__ w1mt,
    const float* __restrict__ w2, const float* __restrict__ s2,
    const unsigned* __restrict__ sel2, float* __restrict__ out) {
    // K-pairs packed contiguously: element (k, c) lives in buf[(k>>1)*P + c].{x,y}
    // -> every WMMA fragment is ONE ds_load_b64 into an aligned VGPR pair.
    __shared__ float2 As2[2 * AS_HALF];   // A tile x2, [kpair][m]
    __shared__ float2 Bs2[2 * BS_HALF];   // B tile x2, [kpair][n^swz]
    __shared__ float  Xs[2 * XS_HALF];    // x chunk x2, [f_local][m]

    const int tid  = threadIdx.x;
    const int lane = tid & 31;
    const int wave = tid >> 5;
    const int wm   = wave & 3;    // 4 waves along M
    const int wn   = wave >> 2;   // 2 waves along N
    const int hh   = lane >> 4;   // half-wave: K pair select
    const int lm   = lane & 15;

    const int m_blk = blockIdx.x & 31;   // M inner -> consecutive blocks reuse w2 slice
    const int n_blk = blockIdx.x >> 5;
    const int m0 = m_blk * BM;
    const int n0 = n_blk * BN;

    const unsigned T2 = sel2[2];

    v8f acc[2][4];
    for (int mt = 0; mt < 2; ++mt)
        for (int nt = 0; nt < 4; ++nt)
            acc[mt][nt] = (v8f){0.f, 0.f, 0.f, 0.f, 0.f, 0.f, 0.f, 0.f};

    // ---- stage x chunk c into Xs[buf]; 128B-coalesced per wave,
    //      conflict-free LDS transpose (banks (c+row)%64 distinct)
    auto fill_xs = [&](int c, int buf) {
        float* xs = Xs + buf * XS_HALF;
        const int fbase = c * KF;
        for (int e = tid; e < BM * KF; e += 256) {
            int row = e >> 5;
            int cc  = e & 31;
            xs[cc * XS_STRIDE + row] =
                x[(size_t)(m0 + row) * F_DIM + fbase + cc];
        }
    };

    // ---- stage s: build A tile (from staged x) and B tile (from HBM, masked)
    auto fill_stage = [&](int s, int buf) {
        const int k0  = s * BK;
        const int fl0 = (s & 7) * 4;              // 4 f per stage within chunk
        const float* xs = Xs + ((s >> 3) & 1) * XS_HALF;
        float2* as = As2 + buf * AS_HALF;
        float2* bs = Bs2 + buf * BS_HALF;
        // A: (k = fsub*8+r, m = i) = relu(x[m0+i, f] * w1m[r, f])
        //    4 x b64 stores per (thread, f); lanes sweep m -> conflict-free
        for (int p = 0; p < 2; ++p) {
            int e = p * 256 + tid;
            int i = e & 127;
            int fsub = e >> 7;                    // 0..3
            float xv = xs[(fl0 + fsub) * XS_STRIDE + i];
            int f = (k0 >> 3) + fsub;
            const float4* wp = (const float4*)(w1mt + f * 8);
            float4 wa = wp[0], wb = wp[1];
            float2* dst = &as[(fsub * 4) * AS_P + i];
            float2 t;
            t.x = fmaxf(xv * wa.x, 0.f); t.y = fmaxf(xv * wa.y, 0.f);
            dst[0 * AS_P] = t;
            t.x = fmaxf(xv * wa.z, 0.f); t.y = fmaxf(xv * wa.w, 0.f);
            dst[1 * AS_P] = t;
            t.x = fmaxf(xv * wb.x, 0.f); t.y = fmaxf(xv * wb.y, 0.f);
            dst[2 * AS_P] = t;
            t.x = fmaxf(xv * wb.z, 0.f); t.y = fmaxf(xv * wb.w, 0.f);
            dst[3 * AS_P] = t;
        }
        // B: bs[klp][n^swz] = mask2 ? w2[n0+n, k0+2klp..+1] : 0
        //    float2 global reads, 128B/wave coalesced along k
        for (int e = tid; e < BN * (BK / 2); e += 256) {
            int n   = e >> 4;
            int klp = e & 15;
            size_t idx = (size_t)(n0 + n) * K_DIM + k0 + 2 * klp;
            float2 wv = *(const float2*)&w2[idx];
            float2 sv = *(const float2*)&s2[idx];
            float2 o;
            o.x = (abs_key(sv.x) >= T2) ? wv.x : 0.0f;
            o.y = (abs_key(sv.y) >= T2) ? wv.y : 0.0f;
            int col = n ^ ((klp & 3) << 2);   // XOR-swizzle: 4-way (not 8) store conflicts
            bs[klp * BS_P + col] = o;
        }
        // gfx1250 global_prefetch_b8 for the stage after this one
        if (k0 + BK < K_DIM && tid < BN) {
            size_t nidx = (size_t)(n0 + tid) * K_DIM + k0 + BK;
            __builtin_prefetch(&w2[nidx], 0, 3);
            __builtin_prefetch(&s2[nidx], 0, 3);
        }
    };

    // ---- compute stage s from buffer: 8 k-steps x 8 tiles of WMMA f32
    auto compute = [&](int buf) {
        const float2* as = As2 + buf * AS_HALF;
        const float2* bs = Bs2 + buf * BS_HALF;
        for (int kk = 0; kk < BK; kk += 4) {
            const int kp = (kk >> 1) + hh;  // ISA: lanes 0-15 K={kk,kk+1}, 16-31 K={kk+2,kk+3}
            v2f a[2], b[4];
            for (int mt = 0; mt < 2; ++mt) {
                int m = wm * 32 + mt * 16 + lm;
                a[mt] = *(const v2f*)&as[kp * AS_P + m];
            }
            const int swz = (kp & 3) << 2;
            for (int nt = 0; nt < 4; ++nt) {
                int n = (wn * 64 + nt * 16 + lm) ^ swz;
                b[nt] = *(const v2f*)&bs[kp * BS_P + n];
            }
            for (int mt = 0; mt < 2; ++mt)
                for (int nt = 0; nt < 4; ++nt)
                    acc[mt][nt] = __builtin_amdgcn_wmma_f32_16x16x4_f32(
                        false, a[mt], false, b[nt],
                        (short)0, acc[mt][nt], false, false);
        }
    };

    // ---- software pipeline: one barrier per stage
    fill_xs(0, 0);
    __syncthreads();
    fill_stage(0, 0);
    for (int s = 0; s < NSTAGE; ++s) {
        __syncthreads();   // fill(s) visible; compute(s-1) done before refills below
        if ((s & 7) == 0 && (s >> 3) + 1 < NCHUNK)
            fill_xs((s >> 3) + 1, ((s >> 3) + 1) & 1);
        if (s + 1 < NSTAGE)
            fill_stage(s + 1, (s + 1) & 1);
        compute(s & 1);
    }

    // --- store C: VGPR v -> M = base + hh*8 + v, N = base + lm
    for (int mt = 0; mt < 2; ++mt)
        for (int nt = 0; nt < 4; ++nt) {
            int mbase = m0 + wm * 32 + mt * 16 + hh * 8;
            int nn    = n0 + wn * 64 + nt * 16 + lm;
            for (int v = 0; v < 8; ++v)
                out[(size_t)(mbase + v) * O_DIM + nn] = acc[mt][nt][v];
        }
}

extern "C" void kernel_launch(void* const* d_in, const int* in_sizes, int n_in,
                              void* d_out, int out_size, void* d_ws, size_t ws_size,
                              hipStream_t stream) {
    (void)in_sizes; (void)n_in; (void)out_size; (void)ws_size;
    const float* x  = (const float*)d_in[0];
    const float* w1 = (const float*)d_in[1];
    const float* w2 = (const float*)d_in[2];
    const float* s1 = (const float*)d_in[3];
    const float* s2 = (const float*)d_in[4];
    float* out = (float*)d_out;

    char* ws = (char*)d_ws;
    unsigned* hist = (unsigned*)ws;
    unsigned* sel1 = (unsigned*)(ws + SEL_OFF);
    unsigned* sel2 = sel1 + 4;
    float* w1mt = (float*)(ws + W1_OFF);

    // ---- exact global median threshold for |scores1| (rank j+1 of 16384)
    const long n1 = (long)R_DIM * F_DIM;
    const unsigned k1 = (unsigned)(n1 / 2) + 1u;
    hipMemsetAsync(hist, 0, HIST_BYTES, stream);
    k_hist_hi<<<64, 256, 0, stream>>>(s1, n1, hist);
    k_scan<<<1, 256, 0, stream>>>(hist, sel1, 0, k1);
    hipMemsetAsync(hist, 0, HIST_BYTES, stream);
    k_hist_lo<<<64, 256, 0, stream>>>(s1, n1, hist, sel1);
    k_scan<<<1, 256, 0, stream>>>(hist, sel1, 1, 0);
    k_mask_w1<<<(R_DIM * F_DIM + 255) / 256, 256, 0, stream>>>(w1, s1, sel1, w1mt);

    // ---- exact global median threshold for |scores2| (rank j+1 of 33554432)
    const long n2 = (long)O_DIM * K_DIM;
    const unsigned k2 = (unsigned)(n2 / 2) + 1u;
    hipMemsetAsync(hist, 0, HIST_BYTES, stream);
    k_hist_hi<<<4096, 256, 0, stream>>>(s2, n2, hist);
    k_scan<<<1, 256, 0, stream>>>(hist, sel2, 0, k2);
    hipMemsetAsync(hist, 0, HIST_BYTES, stream);
    k_hist_lo<<<4096, 256, 0, stream>>>(s2, n2, hist, sel2);
    k_scan<<<1, 256, 0, stream>>>(hist, sel2, 1, 0);

    // ---- fused masked GEMM
    k_gemm<<<(B_ROWS / BM) * (O_DIM / BN), 256, 0, stream>>>(
        x, w1mt, w2, s2, sel2, out);
}